// MultiLabelConstantMarginLoss_73014444032082
// MI455X (gfx1250) — compile-verified
//
#include <hip/hip_runtime.h>

// MultiLabel margin loss:
//   loss = ( sum_{b,k valid} [ sum_j max(0, 1 - x[b,t] + x[b,j]) ] - nValid ) / C
// B=512, C=4096, K=50. Inputs: d_in[0]=float32 x[B,C], d_in[1]=int target[B,K].
// Output: single float32 scalar.

typedef __attribute__((ext_vector_type(2))) float v2f;
typedef __attribute__((ext_vector_type(8))) float v8f;

#define B_DIM 512
#define C_DIM 4096
#define K_DIM 50
#define KPAD  64
#define TPB   256

__global__ __launch_bounds__(TPB) void mlmml_row_kernel(
    const float* __restrict__ x,
    const int*   __restrict__ tgt,
    float*       __restrict__ partial)
{
    __shared__ float srow[C_DIM];   // one full row of x, staged via async LDS DMA
    __shared__ float aval[KPAD];    // a_k = 1 - x[b, t_k]; pads = -1e30 (relu -> 0)
    __shared__ float psum[TPB];     // per-thread partial sums
    __shared__ int   nv;            // number of valid labels in this row

    const int tid = threadIdx.x;
    const int b   = blockIdx.x;
    const float* rowp = x + (size_t)b * C_DIM;

    // ---- Stage row into LDS with CDNA5 async global->LDS copies (ASYNCcnt) ----
    // 256 threads x 16B x 4 iterations = 16 KB = one row.
    #pragma unroll
    for (int it = 0; it < 4; ++it) {
        unsigned lds_off = (unsigned)(unsigned long long)(&srow[0])
                         + (unsigned)(tid * 16 + it * 4096);
        const void* g = (const void*)((const char*)rowp + tid * 16 + it * 4096);
        asm volatile("global_load_async_to_lds_b128 %0, %1, off"
                     :: "v"(lds_off), "v"(g)
                     : "memory");
    }
    if (tid == 0) nv = 0;
    // Each wave waits for its own async transfers, then the block syncs.
    asm volatile("s_wait_asynccnt 0" ::: "memory");
    __syncthreads();

    // ---- Gather a_k = 1 - x[b, t_k] from the LDS-resident row ----
    if (tid < KPAD) {
        float a = -1e30f;                       // pad: relu(a + x) == 0 always
        if (tid < K_DIM) {
            int t = tgt[b * K_DIM + tid];
            if (t > -1) {
                int tc = (t < C_DIM) ? t : (C_DIM - 1);
                a = 1.0f - srow[tc];
                atomicAdd(&nv, 1);
            }
        }
        aval[tid] = a;
    }
    __syncthreads();

    // ---- Main loop: each thread owns 16 strided j values (conflict-free LDS) ----
    float xv[16];
    #pragma unroll
    for (int m = 0; m < 16; ++m) xv[m] = srow[tid + 256 * m];

    float acc = 0.0f;
    #pragma unroll 4
    for (int k = 0; k < KPAD; ++k) {
        const float a = aval[k];               // wave32 LDS broadcast read
        #pragma unroll
        for (int m = 0; m < 16; ++m)
            acc += fmaxf(a + xv[m], 0.0f);
    }
    psum[tid] = acc;
    __syncthreads();

    // ---- Block reduction: 256 -> 32 in VALU, then 32 -> 1 via one WMMA ----
    if (tid < 32) {                            // exactly wave 0: EXEC is all ones
        float s = 0.0f;
        #pragma unroll
        for (int w = 0; w < 8; ++w) s += psum[tid + 32 * w];

        // A (16x4 f32): lane m (0..15) -> A[m][0]=s_m (VGPR0), lanes 16..31 -> A[m][2]=s_{m+16}.
        // With a.y = 0 and B = ones(4x16):  D[m][n] = s_m + s_{m+16}  for all n.
        v2f a;  a.x = s;    a.y = 0.0f;
        v2f bm; bm.x = 1.0f; bm.y = 1.0f;
        v8f c = {};
        v8f d = __builtin_amdgcn_wmma_f32_16x16x4_f32(
            /*neg_a=*/false, a, /*neg_b=*/false, bm,
            /*c_mod=*/(short)0, c, /*reuse_a=*/false, /*reuse_b=*/false);

        // Lane n sums D[r][n] over its 8 VGPRs: lanes 0..15 cover M=0..7, lanes 16..31 cover M=8..15.
        float t = d[0] + d[1] + d[2] + d[3] + d[4] + d[5] + d[6] + d[7];
        t += __shfl_xor(t, 16, 32);            // lane0 now holds full 32-lane sum
        if (tid == 0) partial[b] = t - (float)nv;
    }
}

// Deterministic fixed-order final reduction of the 512 per-row partials.
__global__ __launch_bounds__(256) void mlmml_reduce_kernel(
    const float* __restrict__ partial, float* __restrict__ out)
{
    __shared__ float sm[256];
    const int tid = threadIdx.x;
    sm[tid] = partial[tid] + partial[tid + 256];
    __syncthreads();
    #pragma unroll
    for (int off = 128; off > 0; off >>= 1) {
        if (tid < off) sm[tid] += sm[tid + off];
        __syncthreads();
    }
    if (tid == 0) out[0] = sm[0] * (1.0f / (float)C_DIM);
}

extern "C" void kernel_launch(void* const* d_in, const int* in_sizes, int n_in,
                              void* d_out, int out_size, void* d_ws, size_t ws_size,
                              hipStream_t stream)
{
    const float* x   = (const float*)d_in[0];   // [B, C] float32
    const int*   tgt = (const int*)d_in[1];     // [B, K] labels (-1 = pad)
    float* partial   = (float*)d_ws;            // B floats of scratch
    float* out       = (float*)d_out;           // 1 float

    mlmml_row_kernel<<<B_DIM, TPB, 0, stream>>>(x, tgt, partial);
    mlmml_reduce_kernel<<<1, 256, 0, stream>>>(partial, out);
}